// YoloLoss_63050119905790
// MI455X (gfx1250) — compile-verified
//
#include <hip/hip_runtime.h>
#include <cmath>

typedef __attribute__((ext_vector_type(2))) float v2f;
typedef __attribute__((ext_vector_type(8))) float v8f;

namespace {
constexpr int kThreads = 256;
constexpr int kNBlkObj = 256;              // objness partial blocks per level
constexpr int kHs[3]   = {80, 40, 20};
constexpr int kNumel[3] = {16 * 3 * 80 * 80, 16 * 3 * 40 * 40, 16 * 3 * 20 * 20};
}

// ANCHORS / STRIDES, precomputed in grid units
__constant__ float c_anch[3][3][2] = {
  {{1.25f, 1.625f}, {2.0f, 3.75f}, {4.125f, 2.875f}},
  {{1.875f, 3.8125f}, {3.875f, 2.8125f}, {3.6875f, 7.4375f}},
  {{3.625f, 2.8125f}, {4.875f, 6.1875f}, {11.65625f, 10.1875f}},
};
__constant__ float c_offs[5][2] = {
  {0.f, 0.f}, {0.5f, 0.f}, {0.f, 0.5f}, {-0.5f, 0.f}, {0.f, -0.5f}};

__device__ __forceinline__ float softplusf(float x) {
  return fmaxf(x, 0.f) + log1pf(expf(-fabsf(x)));
}
__device__ __forceinline__ float sigm(float x) { return 1.f / (1.f + expf(-x)); }

__device__ __forceinline__ float ciou_f(float px, float py, float pw, float ph,
                                        float tx, float ty, float tw, float th) {
  const float eps = 1e-7f;
  float b1x1 = px - pw * 0.5f, b1x2 = px + pw * 0.5f;
  float b1y1 = py - ph * 0.5f, b1y2 = py + ph * 0.5f;
  float b2x1 = tx - tw * 0.5f, b2x2 = tx + tw * 0.5f;
  float b2y1 = ty - th * 0.5f, b2y2 = ty + th * 0.5f;
  float iw = fmaxf(fminf(b1x2, b2x2) - fmaxf(b1x1, b2x1), 0.f);
  float ih = fmaxf(fminf(b1y2, b2y2) - fmaxf(b1y1, b2y1), 0.f);
  float inter = iw * ih;
  float uni = pw * ph + tw * th - inter + eps;
  float iou = inter / uni;
  float cw = fmaxf(b1x2, b2x2) - fminf(b1x1, b2x1);
  float ch = fmaxf(b1y2, b2y2) - fminf(b1y1, b2y1);
  float c2 = cw * cw + ch * ch + eps;
  float dx = b2x1 + b2x2 - b1x1 - b1x2;
  float dy = b2y1 + b2y2 - b1y1 - b1y2;
  float rho2 = (dx * dx + dy * dy) * 0.25f;
  float dv = atanf(tw / (th + eps)) - atanf(pw / (ph + eps));
  float v = 0.40528473456935108f * dv * dv;           // 4/pi^2
  float alpha = v / (v - iou + (1.f + eps));          // stop_gradient: same value
  return iou - (rho2 / c2 + v * alpha);
}

__global__ void yolo_zero(float* __restrict__ p, int n) {
  for (int i = blockIdx.x * blockDim.x + threadIdx.x; i < n;
       i += gridDim.x * blockDim.x)
    p[i] = 0.f;
}

// One thread per (offset, anchor, target) candidate. Deterministic: partial
// sums reduced in fixed block order; tobj uses max (order-independent).
__global__ void yolo_match(const float* __restrict__ pred,
                           const float* __restrict__ targets, int nt, int level,
                           int H, int W, float* __restrict__ tobj,
                           float* __restrict__ lbox_part,
                           float* __restrict__ lcls_part,
                           float* __restrict__ cnt_part) {
  const int tid = threadIdx.x;
  const int idx = blockIdx.x * blockDim.x + tid;
  const int total = 5 * 3 * nt;
  float lbox_v = 0.f, lcls_v = 0.f, cnt_v = 0.f;
  if (idx < total) {
    const int o = idx / (3 * nt);
    const int rem = idx - o * 3 * nt;
    const int a = rem / nt;
    const int t = rem - a * nt;
    const float* tg = targets + (size_t)t * 6;
    const float gx = tg[2] * (float)W;
    const float gy = tg[3] * (float)H;
    const float gw = tg[4] * (float)W;
    const float gh = tg[5] * (float)H;
    const float aw = c_anch[level][a][0];
    const float ah = c_anch[level][a][1];
    const float rx = gw / aw, ry = gh / ah;
    const float rmax = fmaxf(fmaxf(rx, 1.f / rx), fmaxf(ry, 1.f / ry));
    bool ok = rmax < 4.0f;  // ANCHOR_T
    if (ok) {
      bool offok;
      if (o == 0) offok = true;
      else if (o == 1) offok = (gx - floorf(gx) < 0.5f) && (gx > 1.0f);
      else if (o == 2) offok = (gy - floorf(gy) < 0.5f) && (gy > 1.0f);
      else if (o == 3) { float gxi = (float)W - gx; offok = (gxi - floorf(gxi) < 0.5f) && (gxi > 1.0f); }
      else             { float gyi = (float)H - gy; offok = (gyi - floorf(gyi) < 0.5f) && (gyi > 1.0f); }
      ok = offok;
    }
    if (ok) {
      cnt_v = 1.f;
      const float offx = c_offs[o][0], offy = c_offs[o][1];
      int gi = (int)(gx - offx); gi = min(max(gi, 0), W - 1);
      int gj = (int)(gy - offy); gj = min(max(gj, 0), H - 1);
      const int b = (int)tg[0];
      const int c = (int)tg[1];
      const float tx = gx - (float)gi, ty = gy - (float)gj;
      const size_t cell = ((size_t)(b * 3 + a) * H + gj) * W + gi;
      const float* ps = pred + cell * 85;
      const float px = sigm(ps[0]) * 2.f - 0.5f;
      const float py = sigm(ps[1]) * 2.f - 0.5f;
      const float sw = sigm(ps[2]) * 2.f;
      const float sh = sigm(ps[3]) * 2.f;
      const float pw = sw * sw * aw;
      const float ph = sh * sh * ah;
      const float iou = ciou_f(px, py, pw, ph, tx, ty, gw, gh);
      lbox_v = 1.f - iou;
      atomicMax((unsigned int*)&tobj[cell], __float_as_uint(fmaxf(iou, 0.f)));
      // sum_j bce(x_j, y_j) = sum_j softplus(x_j) - x_c   (CP=1, CN=0)
      float cls = 0.f;
      for (int j = 0; j < 80; ++j) cls += softplusf(ps[5 + j]);
      lcls_v = cls - ps[5 + c];
    }
  }
  __shared__ float sb[kThreads], sc[kThreads], sn[kThreads];
  sb[tid] = lbox_v; sc[tid] = lcls_v; sn[tid] = cnt_v;
  __syncthreads();
  for (int s = kThreads / 2; s > 0; s >>= 1) {
    if (tid < s) { sb[tid] += sb[tid + s]; sc[tid] += sc[tid + s]; sn[tid] += sn[tid + s]; }
    __syncthreads();
  }
  if (tid == 0) {
    lbox_part[blockIdx.x] = sb[0];
    lcls_part[blockIdx.x] = sc[0];
    cnt_part[blockIdx.x]  = sn[0];
  }
}

// Dense objectness BCE over channel 4: bce(x,y) = softplus(x) - x*y
__global__ void yolo_obj(const float* __restrict__ pred,
                         const float* __restrict__ tobj, int numel,
                         float* __restrict__ part) {
  const int tid = threadIdx.x;
  const int stride = gridDim.x * blockDim.x;
  float acc = 0.f;
  for (int n = blockIdx.x * blockDim.x + tid; n < numel; n += stride) {
    __builtin_prefetch(&pred[(size_t)(n + stride) * 85 + 4], 0, 1);  // global_prefetch_b8
    const float x = pred[(size_t)n * 85 + 4];
    const float y = tobj[n];
    acc += softplusf(x) - x * y;
  }
  __shared__ float sd[kThreads];
  sd[tid] = acc;
  __syncthreads();
  for (int s = kThreads / 2; s > 0; s >>= 1) {
    if (tid < s) sd[tid] += sd[tid + s];
    __syncthreads();
  }
  if (tid == 0) part[blockIdx.x] = sd[0];
}

// Wave32 sum via V_WMMA_F32_16X16X4_F32 with ones-B (layout-proof):
// A[m,0]=v[lane m], A[m,2]=v[lane m+16]  ->  D[m,n] = v[m]+v[m+16].
// Sum of the 8 D VGPRs gives half-sums per lane half; shfl_xor 16 completes.
__device__ __forceinline__ float wave_sum32(float v) {
  v2f a; a.x = v;  a.y = 0.f;
  v2f b; b.x = 1.f; b.y = 1.f;
  v8f c = {};
  v8f d = __builtin_amdgcn_wmma_f32_16x16x4_f32(false, a, false, b, (short)0, c,
                                                false, false);
  float s = d[0] + d[1] + d[2] + d[3] + d[4] + d[5] + d[6] + d[7];
  s += __shfl_xor(s, 16, 32);
  return s;
}

__global__ void yolo_final(const float* __restrict__ lboxp,
                           const float* __restrict__ lclsp,
                           const float* __restrict__ cntp,
                           const float* __restrict__ lobjp, int nbB,
                           float* __restrict__ out) {
  const int lane = threadIdx.x;
  const float bal[3] = {4.f, 1.f, 0.4f};
  const float numel[3] = {307200.f, 76800.f, 19200.f};
  float lbox = 0.f, lobj = 0.f, lcls = 0.f;
  for (int i = 0; i < 3; ++i) {
    float sb = 0.f, sc = 0.f, sn = 0.f, so = 0.f;
    for (int j = lane; j < nbB; j += 32) {
      sb += lboxp[i * nbB + j];
      sc += lclsp[i * nbB + j];
      sn += cntp[i * nbB + j];
    }
    for (int j = lane; j < kNBlkObj; j += 32) so += lobjp[i * kNBlkObj + j];
    sb = wave_sum32(sb);
    sc = wave_sum32(sc);
    sn = wave_sum32(sn);
    so = wave_sum32(so);
    const float n = fmaxf(sn, 1.f);
    lbox += sb / n;
    lcls += sc / (n * 80.f);
    lobj += (so / numel[i]) * bal[i];
  }
  if (lane == 0) {
    out[0] = lbox * 0.8f;   // BOX_F * 3/NL * B
    out[1] = lobj * 11.2f;  // OBJ_F * 3/NL * B
    out[2] = lcls * 4.8f;   // CLS_F * NC/80 * 3/NL * B
  }
}

extern "C" void kernel_launch(void* const* d_in, const int* in_sizes, int n_in,
                              void* d_out, int out_size, void* d_ws,
                              size_t ws_size, hipStream_t stream) {
  (void)n_in; (void)out_size; (void)ws_size;
  const float* preds[3] = {(const float*)d_in[0], (const float*)d_in[1],
                           (const float*)d_in[2]};
  const float* targets = (const float*)d_in[3];
  const int nt = in_sizes[3] / 6;

  float* ws = (float*)d_ws;
  const int totalCand = 15 * nt;
  const int nbB = (totalCand + kThreads - 1) / kThreads;

  size_t o_tobj0 = 0;
  size_t o_tobj1 = o_tobj0 + (size_t)kNumel[0];
  size_t o_tobj2 = o_tobj1 + (size_t)kNumel[1];
  size_t base    = o_tobj2 + (size_t)kNumel[2];
  float* tobj[3] = {ws + o_tobj0, ws + o_tobj1, ws + o_tobj2};
  float* lboxp = ws + base;
  float* lclsp = lboxp + 3 * (size_t)nbB;
  float* cntp  = lclsp + 3 * (size_t)nbB;
  float* lobjp = cntp + 3 * (size_t)nbB;
  const size_t total_ws = base + 9 * (size_t)nbB + 3 * (size_t)kNBlkObj;

  // tobj + partials must be zero every call (harness does not re-poison)
  yolo_zero<<<256, kThreads, 0, stream>>>(ws, (int)total_ws);

  for (int i = 0; i < 3; ++i)
    yolo_match<<<nbB, kThreads, 0, stream>>>(
        preds[i], targets, nt, i, kHs[i], kHs[i], tobj[i], lboxp + (size_t)i * nbB,
        lclsp + (size_t)i * nbB, cntp + (size_t)i * nbB);

  for (int i = 0; i < 3; ++i)
    yolo_obj<<<kNBlkObj, kThreads, 0, stream>>>(preds[i], tobj[i], kNumel[i],
                                                lobjp + (size_t)i * kNBlkObj);

  yolo_final<<<1, 32, 0, stream>>>(lboxp, lclsp, cntp, lobjp, nbB,
                                   (float*)d_out);
}